// Loss_55533927137965
// MI455X (gfx1250) — compile-verified
//
#include <hip/hip_runtime.h>
#include <math.h>

// ---------------------------------------------------------------------------
// recon = mean_b sum |x - x_out|   : pure HBM-streaming reduction (~100.7 MB)
//         -> bandwidth-bound; 23.3 TB/s floor ~4.3 us. b128 NT loads,
//            2048 blocks x 256 thr (16K wave32s), 6 float4 iters/thread.
// kld   : tiny per-row computation on (256,7)/(256,32) + disc_pos(10)
// out   = { recon, kld, recon + kld }
// Wave-level reduction runs on the matrix pipe (v_wmma_f32_16x16x4_f32 vs
// an all-ones B), which is free in a streaming kernel.
// ---------------------------------------------------------------------------

typedef __attribute__((ext_vector_type(2))) float v2f;  // A/B of 16x16x4 f32 WMMA
typedef __attribute__((ext_vector_type(4))) float v4f;  // 128-bit vector loads
typedef __attribute__((ext_vector_type(8))) float v8f;  // C/D of 16x16 f32 WMMA

#define TPB 256
#define NBLK 2048

// Wave32 full reduction using the matrix pipe:
// A(16x4) holds lane partials (VGPR0 = K0 for lanes 0-15 / K2 for lanes 16-31,
// VGPR1 = 0), B = ones(4x16)  =>  D[m][n] = p[m] + p[m+16] for every n.
// Per-lane sum of the 8 D VGPRs gives rows 0-7 (lanes 0-15) / rows 8-15
// (lanes 16-31); one shfl_xor(16) completes the 32-lane sum in every lane.
__device__ __forceinline__ float wave_reduce_wmma(float p) {
    v2f a; a.x = p;    a.y = 0.0f;
    v2f b; b.x = 1.0f; b.y = 1.0f;
    v8f c = {};
    c = __builtin_amdgcn_wmma_f32_16x16x4_f32(
            /*neg_a=*/false, a, /*neg_b=*/false, b,
            /*c_mod=*/(short)0, c, /*reuse_a=*/false, /*reuse_b=*/false);
    float s = c[0] + c[1] + c[2] + c[3] + c[4] + c[5] + c[6] + c[7];
    s += __shfl_xor(s, 16, 32);
    return s;
}

__global__ void recon_partial_kernel(const float* __restrict__ x,
                                     const float* __restrict__ xo,
                                     float* __restrict__ partial,
                                     int n4, int n_total) {
    const int tid    = blockIdx.x * blockDim.x + threadIdx.x;
    const int stride = gridDim.x * blockDim.x;

    const v4f* X  = (const v4f*)x;
    const v4f* XO = (const v4f*)xo;

    float s = 0.0f;
    for (int i = tid; i < n4; i += stride) {
        v4f a = __builtin_nontemporal_load(&X[i]);
        v4f b = __builtin_nontemporal_load(&XO[i]);
        s += fabsf(a.x - b.x) + fabsf(a.y - b.y) +
             fabsf(a.z - b.z) + fabsf(a.w - b.w);
    }
    // scalar tail (none for this problem size, kept for robustness)
    if (tid == 0) {
        for (int i = n4 * 4; i < n_total; ++i) s += fabsf(x[i] - xo[i]);
    }

    // full EXEC here (loop reconverged) -> WMMA-based wave reduction
    float wsum = wave_reduce_wmma(s);

    __shared__ float lds[TPB / 32];
    const int wave = threadIdx.x >> 5;
    const int lane = threadIdx.x & 31;
    if (lane == 0) lds[wave] = wsum;
    __syncthreads();
    if (threadIdx.x == 0) {
        float t = 0.0f;
        for (int w = 0; w < TPB / 32; ++w) t += lds[w];
        partial[blockIdx.x] = t;   // deterministic per-block partial (no atomics)
    }
}

__global__ void finalize_kernel(const float* __restrict__ partial, int nPartial,
                                const float* __restrict__ y,
                                const float* __restrict__ mu,
                                const float* __restrict__ disc_pos,
                                float* __restrict__ out) {
    __shared__ float red_r[TPB];
    __shared__ float red_k[TPB];
    __shared__ float dp[10];

    const int t = threadIdx.x;
    if (t < 10) dp[t] = disc_pos[t];
    __syncthreads();

    // deterministic fixed-order sum of the recon block partials
    float rs = 0.0f;
    for (int k = t; k < nPartial; k += TPB) rs += partial[k];

    // kld terms for batch row t  (n_disc = 3, LIN = 4, y row width 7)
    float ks = 0.0f;
    {
        const float* yr  = y  + t * 7;
        const float* mur = mu + t * 32;
        // discrete dims
        for (int i = 0; i < 3; ++i) {
            float m  = mur[i];
            float yv = yr[i];
            float term;
            if (isnan(yv)) {                       // unlabeled: min_n (m - dp[n])^2
                float best = 3.402823466e+38f;
                for (int n = 0; n < 10; ++n) {
                    float d = m - dp[n];
                    best = fminf(best, d * d);
                }
                term = best;
            } else {                               // labeled: (m - dp[clip(y,0,9)])^2
                float yc  = fminf(fmaxf(yv, 0.0f), 9.0f);
                int   idx = (int)yc;
                float d   = m - dp[idx];
                term = d * d;
            }
            ks += term;
        }
        // linear dims
        for (int j = 0; j < 4; ++j) {
            float m  = mur[3 + j];
            float yv = yr[3 + j];
            float term;
            if (!isnan(yv)) {
                float d = m - yv;
                term = d * d;
            } else {
                float d = fmaxf(fabsf(m) - 10.0f, 0.0f);  // relu(|m|-10)^2
                term = d * d;
            }
            ks += term;
        }
    }

    red_r[t] = rs;
    red_k[t] = ks;
    __syncthreads();
    for (int off = TPB / 2; off > 0; off >>= 1) {
        if (t < off) {
            red_r[t] += red_r[t + off];
            red_k[t] += red_k[t + off];
        }
        __syncthreads();
    }
    if (t == 0) {
        const float invB  = 1.0f / 256.0f;
        float recon = red_r[0] * invB;   // mean over batch of per-sample sums
        float kld   = red_k[0] * invB;   // kld_disc(sum/B) + kld_lin(sum/B)
        out[0] = recon;
        out[1] = kld;
        out[2] = recon + kld;
    }
}

extern "C" void kernel_launch(void* const* d_in, const int* in_sizes, int n_in,
                              void* d_out, int out_size, void* d_ws, size_t ws_size,
                              hipStream_t stream) {
    const float* x  = (const float*)d_in[0];   // (256,3,128,128)
    const float* xo = (const float*)d_in[1];   // (256,3,128,128)
    const float* y  = (const float*)d_in[2];   // (256,7)
    const float* mu = (const float*)d_in[3];   // (256,32)
    const float* dp = (const float*)d_in[4];   // (10,)
    float* out      = (float*)d_out;           // 3 floats
    float* partial  = (float*)d_ws;            // NBLK floats of scratch

    const int n  = in_sizes[0];                // 12,582,912
    const int n4 = n / 4;                      // 3,145,728 float4s

    recon_partial_kernel<<<NBLK, TPB, 0, stream>>>(x, xo, partial, n4, n);
    finalize_kernel<<<1, TPB, 0, stream>>>(partial, NBLK, y, mu, dp, out);
}